// RCBLayer_4329327035139
// MI455X (gfx1250) — compile-verified
//
#include <hip/hip_runtime.h>
#include <hip/hip_bf16.h>

typedef __attribute__((ext_vector_type(2))) float v2f;
typedef __attribute__((ext_vector_type(8))) float v8f;

#define BATCH 8
#define CHAN  512
#define HW    4096      // 64*64
#define WIDTH 64
#define PW    65        // padded width/height for LDS (zero pad row/col 64)
#define NPATCH 1024     // 32*32 patches per batch

// ---------------------------------------------------------------------------
// Phase 1: two 1x1 convs + ReLU -> recon (from spade_fm,w1,b1), refmap (x,w2,b2)
// Block: (64 pixels, 4 channel groups of 128). Grid: 32768/64 = 512 blocks.
// Each wave (32 lanes, same channel group) issues coalesced 128B loads per c.
// ---------------------------------------------------------------------------
__global__ void __launch_bounds__(256)
rcb_phase1(const float* __restrict__ fm, const float* __restrict__ x,
           const float* __restrict__ w1, const float* __restrict__ b1,
           const float* __restrict__ w2, const float* __restrict__ b2,
           float* __restrict__ recon, float* __restrict__ refmap) {
    __shared__ float sw1[CHAN];
    __shared__ float sw2[CHAN];
    __shared__ float part1[4][64];
    __shared__ float part2[4][64];

    const int tid = threadIdx.y * 64 + threadIdx.x;
    for (int i = tid; i < CHAN; i += 256) {
        sw1[i] = w1[i];
        sw2[i] = w2[i];
    }
    __syncthreads();

    const int pixBase = blockIdx.x * 64;          // 64 divides 4096 -> one batch per block
    const int b       = pixBase >> 12;
    const int inb     = (pixBase & (HW - 1)) + threadIdx.x;  // pixel within batch

    const float* fmb = fm + (size_t)b * CHAN * HW + inb;
    const float* xb  = x  + (size_t)b * CHAN * HW + inb;

    const int c0 = threadIdx.y * 128;
    float s1 = 0.f, s2 = 0.f;
#pragma unroll 4
    for (int c = 0; c < 128; ++c) {
        const int cc = c0 + c;
        s1 += fmb[(size_t)cc * HW] * sw1[cc];
        s2 += xb [(size_t)cc * HW] * sw2[cc];
    }
    part1[threadIdx.y][threadIdx.x] = s1;
    part2[threadIdx.y][threadIdx.x] = s2;
    __syncthreads();

    if (threadIdx.y == 0) {
        float t1 = part1[0][threadIdx.x] + part1[1][threadIdx.x] +
                   part1[2][threadIdx.x] + part1[3][threadIdx.x] + b1[0];
        float t2 = part2[0][threadIdx.x] + part2[1][threadIdx.x] +
                   part2[2][threadIdx.x] + part2[3][threadIdx.x] + b2[0];
        recon [b * HW + inb] = fmaxf(t1, 0.f);
        refmap[b * HW + inb] = fmaxf(t2, 0.f);
    }
}

// ---------------------------------------------------------------------------
// Phase 2: per-batch correlation GEMM (1024x4) @ (4x4096) via
// V_WMMA_F32_16X16X4_F32, fused running argmax (first-index tie-break),
// writes offsets[b*1024 + l] = argmax_n yi[b,l,n].
// Block: 256 threads = 8 waves; wave w owns M-tile m = (blk%8)*8 + w.
// Grid: 8 batches * 8 blocks = 64 blocks.
// recon[b] staged in LDS as a 65x65 tile with the reference's zero padding
// (bottom row / right col) baked in, so the B-fragment build is two adjacent
// unconditional LDS reads -> one ds_load_b64, no EXEC-divergent branches.
// ---------------------------------------------------------------------------
__global__ void __launch_bounds__(256)
rcb_phase2(const float* __restrict__ recon, const float* __restrict__ refmap,
           int* __restrict__ offsets) {
    __shared__ float sre[PW * PW];   // 65x65 padded recon map (post-ReLU)

    const int b  = blockIdx.x >> 3;
    const int mg = blockIdx.x & 7;
    const int tid = threadIdx.x;

    // interior
    for (int i = tid; i < HW; i += 256) {
        const int p = i >> 6, q = i & 63;
        sre[p * PW + q] = recon[b * HW + i];
    }
    // zero padding: row 64 (65 cells incl. corner) and col 64 (64 cells)
    if (tid < PW)    sre[64 * PW + tid] = 0.f;
    if (tid < WIDTH) sre[tid * PW + 64] = 0.f;
    __syncthreads();

    const int lane = tid & 31;
    const int wave = tid >> 5;
    const int m    = mg * 8 + wave;           // M-tile index (rows l = m*16 .. m*16+15)
    const int lcol = lane & 15;
    const int hi   = (lane >= 16) ? 1 : 0;    // lanes 16-31 carry K=2,3 (di=1)
    const int di   = hi;

    // A fragment (16x4 f32): lane L<16 -> row m*16+L, K=0,1 ; lane>=16 -> K=2,3
    // P[l, k=(di,dj)] = refmap[b, 2*li+di, 2*lj+dj],  l=(li,lj) row-major 32x32
    const int row = m * 16 + lcol;
    const int li  = row >> 5;
    const int lj  = row & 31;
    const float* rf = refmap + (size_t)b * HW;
    v2f A;
    A.x = rf[(2 * li + di) * WIDTH + 2 * lj + 0];
    A.y = rf[(2 * li + di) * WIDTH + 2 * lj + 1];

    float best[8];
    int   bidx[8];
#pragma unroll
    for (int r = 0; r < 8; ++r) {
        best[r] = -__builtin_huge_valf();
        bidx[r] = 0;
    }

    // Loop over 256 N-tiles of 16 columns.  n = p*64+q flat spatial index.
#pragma unroll 4
    for (int t = 0; t < 256; ++t) {
        const int n = t * 16 + lcol;
        const int p = n >> 6;
        const int q = n & 63;
        // B fragment (4x16 f32): W[k=(di,dj), n] = padded recon[p+di, q+dj]
        const int base = (p + di) * PW + q;   // always in-bounds of 65x65 tile
        v2f B;
        B.x = sre[base];
        B.y = sre[base + 1];

        v8f C = {};
        v8f D = __builtin_amdgcn_wmma_f32_16x16x4_f32(
            false, A, false, B, (short)0, C, false, false);

        // D layout: VGPR r -> (M = r, N = lane) for lanes 0-15,
        //                     (M = 8+r, N = lane-16) for lanes 16-31.
        // Per-lane running argmax; n strictly increases => '>' keeps first hit.
#pragma unroll
        for (int r = 0; r < 8; ++r) {
            const float v = D[r];
            if (v > best[r]) { best[r] = v; bidx[r] = n; }
        }
    }

    // Cross-lane reduce within each 16-lane half (xor masks stay in-half).
#pragma unroll
    for (int r = 0; r < 8; ++r) {
        float v = best[r];
        int   n = bidx[r];
#pragma unroll
        for (int mask = 8; mask >= 1; mask >>= 1) {
            const float ov = __shfl_xor(v, mask, 32);
            const int   on = __shfl_xor(n, mask, 32);
            if (ov > v || (ov == v && on < n)) { v = ov; n = on; }
        }
        best[r] = v;
        bidx[r] = n;
    }

    if (lcol == 0) {
        const int base = b * NPATCH + m * 16 + (hi ? 8 : 0);
#pragma unroll
        for (int r = 0; r < 8; ++r)
            offsets[base + r] = bidx[r];
    }
}

// ---------------------------------------------------------------------------
// Phase 3: idx = offset>>2; copy 2x2 patch (idx>>5, idx&31) of recon into
// output patch l.  8*1024 threads.
// ---------------------------------------------------------------------------
__global__ void __launch_bounds__(256)
rcb_phase3(const float* __restrict__ recon, const int* __restrict__ offsets,
           float* __restrict__ out) {
    const int g = blockIdx.x * blockDim.x + threadIdx.x;  // 0..8191
    const int b = g >> 10;
    const int l = g & (NPATCH - 1);
    const int off = offsets[g];
    const int idx = off >> 2;
    const int pr = idx >> 5, pc = idx & 31;
    const int li = l >> 5,   lj = l & 31;
    const float* rb = recon + b * HW;
    float* ob = out + b * HW;
#pragma unroll
    for (int dii = 0; dii < 2; ++dii)
#pragma unroll
        for (int dj = 0; dj < 2; ++dj)
            ob[(2 * li + dii) * WIDTH + 2 * lj + dj] =
                rb[(2 * pr + dii) * WIDTH + 2 * pc + dj];
}

extern "C" void kernel_launch(void* const* d_in, const int* in_sizes, int n_in,
                              void* d_out, int out_size, void* d_ws, size_t ws_size,
                              hipStream_t stream) {
    (void)in_sizes; (void)n_in; (void)out_size; (void)ws_size;
    const float* fm = (const float*)d_in[0];   // spade_fm (8,512,64,64)
    const float* x  = (const float*)d_in[1];   // x        (8,512,64,64)
    const float* w1 = (const float*)d_in[2];   // (1,512)
    const float* b1 = (const float*)d_in[3];   // (1,)
    const float* w2 = (const float*)d_in[4];   // (1,512)
    const float* b2 = (const float*)d_in[5];   // (1,)
    float* out = (float*)d_out;                // (8,1,64,64)

    float* recon   = (float*)d_ws;             // 8*4096 floats
    float* refmap  = recon + BATCH * HW;       // 8*4096 floats
    int*   offsets = (int*)(refmap + BATCH * HW);  // 8*1024 ints

    rcb_phase1<<<512, dim3(64, 4), 0, stream>>>(fm, x, w1, b1, w2, b2, recon, refmap);
    rcb_phase2<<<64, 256, 0, stream>>>(recon, refmap, offsets);
    rcb_phase3<<<32, 256, 0, stream>>>(recon, offsets, out);
}